// WaveletTransform_5016521801948
// MI455X (gfx1250) — compile-verified
//
#include <hip/hip_runtime.h>
#include <hip/hip_bf16.h>

// db4 wavelet decomposition (depthwise stride-2 conv1d, K=8, two filter banks)
// via V_WMMA_F32_16X16X4_F32 on gfx1250 (MI455X).
//
// Roofline: 128 MiB in + 128 MiB out @ 23.3 TB/s => ~11.5 us floor; the
// ~0.5 GFLOP of filter math runs on the WMMA pipe, costing no VALU issue slots.
// Per 16-output tile per wave: 2 global_load_b64 per lane (overlapping lanes
// share cachelines -> WGP$/L2 absorb redundancy, HBM sees ~unique bytes),
// 2 chained WMMAs, 1 coalesced b32 store per lane (lanes 0-15 = approx row M=0,
// lanes 16-31 = detail row M=8 of the D matrix).
//
// readfirstlane makes wave/row/col compiler-provably uniform: boundary checks
// become s_cbranch (EXEC stays all-ones for WMMA) and all memory ops use the
// saddr + 32-bit voffset form with immediate tile offsets.

typedef __attribute__((ext_vector_type(2))) float v2f;
typedef __attribute__((ext_vector_type(8))) float v8f;

#define L_IN           16384
#define L_OUT          8192
#define ROWS           2048          // B*C = 32*64
#define TILES_PER_ROW  512           // L_OUT / 16
#define TILES_PER_WAVE 16
#define OUT_HALF       16777216      // ROWS * L_OUT, fits in int

// Zero-padded load, branchless (clamp + select) so EXEC is never perturbed.
__device__ __forceinline__ float ldx(const float* __restrict__ r, int i) {
    int ic = i < 0 ? 0 : (i > (L_IN - 1) ? (L_IN - 1) : i);
    float v = r[ic];
    return (i == ic) ? v : 0.0f;
}

extern "C" __global__ __launch_bounds__(256)
void dwt_db4_wmma_kernel(const float* __restrict__ x, float* __restrict__ out) {
    const int lane = threadIdx.x & 31;          // wave32
    // Scalarize the wave id: threadIdx.x>>5 is wave-uniform but the compiler
    // can't prove it; readfirstlane pins it (and everything derived) to SGPRs.
    const int wave = __builtin_amdgcn_readfirstlane(
        (int)(blockIdx.x * 8u) + (int)(threadIdx.x >> 5));

    const int n   = lane & 15;                  // B column / A row M / pos in tile
    const int kk  = lane >> 4;                  // which K-pair this lane half holds
    const bool isH = (n < 8);                   // A rows 0-7 = H taps, 8-15 = G taps
    const bool hiK = (kk != 0);

    // Cross-correlation taps = reversed filters: taps[k] = FILT[7-k].
    // A layout (ISA 16x4 f32): lane m -> {K0,K1}; lane m+16 -> {K2,K3}.
    v2f a1, a2;
    a1.x = isH ? (hiK ? -0.63088076792959f   : -0.230377813308896f)
               : (hiK ? -0.030841381835987f  :  0.010597401784997f);
    a1.y = isH ? (hiK ?  0.027983769416984f  :  0.714846570552542f)
               : (hiK ?  0.187034811718881f  : -0.032883011666983f);
    a2.x = isH ? (hiK ? -0.032883011666983f  :  0.187034811718881f)
               : (hiK ? -0.714846570552542f  : -0.027983769416984f);
    a2.y = isH ? (hiK ? -0.010597401784997f  :  0.030841381835987f)
               : (hiK ? -0.230377813308896f  : -0.63088076792959f);

    const int wt   = wave * TILES_PER_WAVE;     // 16 consecutive tiles in one row
    const int row  = wt >> 9;                   // / TILES_PER_ROW
    const int col0 = wt & (TILES_PER_ROW - 1);

    const float* __restrict__ xr = x + (size_t)row * L_IN;   // uniform base (SGPR)
    // 32-bit per-lane output index: lanes 0-15 -> approx, 16-31 -> detail.
    const int obase = (lane >= 16 ? OUT_HALF : 0) + row * L_OUT + n;
    const int d = 2 * n + 2 * kk;               // even per-lane window offset

    #pragma unroll
    for (int t = 0; t < TILES_PER_WAVE; ++t) {
        const int col  = col0 + t;              // uniform (SGPR) -> scalar branch
        const int base = col * 32 + d;

        // B layout (4x16 f32): lane n -> {K0,K1}, lane n+16 -> {K2,K3}:
        //   b1 = x[32c + 2n - 3 + 2kk + {0,1}]   (taps 0-3 window)
        //   b2 = x[32c + 2n + 1 + 2kk + {0,1}]   (taps 4-7 window)
        v2f b1, b2;
        if (__builtin_expect(col != 0 && col != (TILES_PER_ROW - 1), 1)) {
            b1.x = xr[base - 3]; b1.y = xr[base - 2];
            b2.x = xr[base + 1]; b2.y = xr[base + 2];
        } else {                                 // scalar-branch boundary path
            b1.x = ldx(xr, base - 3); b1.y = ldx(xr, base - 2);
            b2.x = ldx(xr, base + 1); b2.y = ldx(xr, base + 2);
        }

        v8f c = {0.f, 0.f, 0.f, 0.f, 0.f, 0.f, 0.f, 0.f};
        c = __builtin_amdgcn_wmma_f32_16x16x4_f32(
                /*neg_a=*/false, a1, /*neg_b=*/false, b1,
                /*c_mod=*/(short)0, c, /*reuse_a=*/false, /*reuse_b=*/false);
        c = __builtin_amdgcn_wmma_f32_16x16x4_f32(
                false, a2, false, b2, (short)0, c, false, false);

        // D VGPR0: lanes 0-15 = approx[col*16+n], lanes 16-31 = detail[col*16+n]
        out[obase + col * 16] = c[0];
    }
}

extern "C" void kernel_launch(void* const* d_in, const int* in_sizes, int n_in,
                              void* d_out, int out_size, void* d_ws, size_t ws_size,
                              hipStream_t stream) {
    (void)in_sizes; (void)n_in; (void)out_size; (void)d_ws; (void)ws_size;
    const float* x = (const float*)d_in[0];
    float* out = (float*)d_out;

    // 2048 rows * 512 tiles / 16 tiles-per-wave = 65536 waves; 8 waves/block.
    dim3 grid(8192), block(256);
    hipLaunchKernelGGL(dwt_db4_wmma_kernel, grid, block, 0, stream, x, out);
}